// LennardJones_24610162606258
// MI455X (gfx1250) — compile-verified
//
#include <hip/hip_runtime.h>
#include <hip/hip_bf16.h>

typedef float v2f __attribute__((ext_vector_type(2)));
typedef float v8f __attribute__((ext_vector_type(8)));
typedef long long v2ll __attribute__((ext_vector_type(2)));

struct __align__(16) LLPair { long long i, j; };

// ---------------------------------------------------------------------------
// Wave32 + block reduction: one V_WMMA_F32_16X16X4_F32 does the 32-lane sum
// (A = partials striped 16x4, B = ones), then LDS across waves, one atomic.
// EXEC must be all-ones here: callers invoke this convergently.
// ---------------------------------------------------------------------------
__device__ __forceinline__ void block_reduce_add(float acc, float* out)
{
    v2f a; a.x = acc;  a.y = 0.0f;
    v2f b; b.x = 1.0f; b.y = 1.0f;
    v8f c = {};
    v8f d8 = __builtin_amdgcn_wmma_f32_16x16x4_f32(
        false, a, false, b, (short)0, c, false, false);

    // lanes 0-15 end with sum(r[0..7]), lanes 16-31 with sum(r[8..15])
    float wsum = d8[0] + d8[1] + d8[2] + d8[3] + d8[4] + d8[5] + d8[6] + d8[7];
    wsum += __shfl_xor(wsum, 16, 32);

    __shared__ float red[8];
    const int lane = threadIdx.x & 31;
    const int wave = threadIdx.x >> 5;
    if (lane == 0) red[wave] = wsum;
    __syncthreads();
    if (threadIdx.x == 0) {
        float s = 0.0f;
        const int nw = blockDim.x >> 5;
        for (int w = 0; w < nw; ++w) s += red[w];
        atomicAdd(out, s);
    }
}

// ---------------------------------------------------------------------------
// Box inverse (adjugate/det) -- loop invariant, computed once per thread.
// ---------------------------------------------------------------------------
struct BoxMats {
    float b00,b01,b02,b10,b11,b12,b20,b21,b22;
    float i00,i01,i02,i10,i11,i12,i20,i21,i22;
};

__device__ __forceinline__ BoxMats load_box(const float* __restrict__ box)
{
    BoxMats m;
    m.b00 = box[0]; m.b01 = box[1]; m.b02 = box[2];
    m.b10 = box[3]; m.b11 = box[4]; m.b12 = box[5];
    m.b20 = box[6]; m.b21 = box[7]; m.b22 = box[8];
    const float det = m.b00*(m.b11*m.b22 - m.b12*m.b21)
                    - m.b01*(m.b10*m.b22 - m.b12*m.b20)
                    + m.b02*(m.b10*m.b21 - m.b11*m.b20);
    const float inv = 1.0f / det;
    m.i00 =  (m.b11*m.b22 - m.b12*m.b21) * inv;
    m.i01 = -(m.b01*m.b22 - m.b02*m.b21) * inv;
    m.i02 =  (m.b01*m.b12 - m.b02*m.b11) * inv;
    m.i10 = -(m.b10*m.b22 - m.b12*m.b20) * inv;
    m.i11 =  (m.b00*m.b22 - m.b02*m.b20) * inv;
    m.i12 = -(m.b00*m.b12 - m.b02*m.b10) * inv;
    m.i20 =  (m.b10*m.b21 - m.b11*m.b20) * inv;
    m.i21 = -(m.b00*m.b21 - m.b01*m.b20) * inv;
    m.i22 =  (m.b00*m.b11 - m.b01*m.b10) * inv;
    return m;
}

// Per-pair energy from packed records (x, y, z, 0.5*sigma | sqrt(eps))
__device__ __forceinline__ float pair_energy_packed(
    const float* __restrict__ rec, long long iL, long long jL,
    const BoxMats& m, float cutoff)
{
    const size_t ia = (size_t)iL, ja = (size_t)jL;
    const float4 ri = ((const float4*)rec)[2*ia];   // b128, 32B-aligned record
    const float  si = rec[8*ia + 4];                // b32, same 64B line
    const float4 rj = ((const float4*)rec)[2*ja];
    const float  sj = rec[8*ja + 4];

    const float dx = ri.x - rj.x, dy = ri.y - rj.y, dz = ri.z - rj.z;

    // ds = dr @ box_inv, minimum image
    float sx = dx*m.i00 + dy*m.i10 + dz*m.i20;
    float sy = dx*m.i01 + dy*m.i11 + dz*m.i21;
    float sz = dx*m.i02 + dy*m.i12 + dz*m.i22;
    sx -= floorf(sx + 0.5f);
    sy -= floorf(sy + 0.5f);
    sz -= floorf(sz + 0.5f);

    // dr_pbc = ds @ box
    const float px = sx*m.b00 + sy*m.b10 + sz*m.b20;
    const float py = sx*m.b01 + sy*m.b11 + sz*m.b21;
    const float pz = sx*m.b02 + sy*m.b12 + sz*m.b22;

    const float d2 = px*px + py*py + pz*pz;
    const float d  = sqrtf(d2);
    const float dsafe = (d > 0.0f) ? d : 1.0f;

    const float sg = ri.w + rj.w;   // 0.5*(sigma_i + sigma_j)
    const float ep = si * sj;       // sqrt(eps_i)*sqrt(eps_j)

    const float s  = sg / dsafe;
    const float s2 = s * s;
    const float t  = s2 * s2 * s2;
    const float e  = 4.0f * ep * t * (t - 1.0f);

    return (d <= cutoff) ? e : 0.0f;
}

// ---------------------------------------------------------------------------
// Pack each atom into a 32B-aligned record: (x, y, z, 0.5*sigma, sqrt(eps),..)
// ---------------------------------------------------------------------------
__global__ void __launch_bounds__(256)
pack_atoms_kernel(const float* __restrict__ coords,
                  const float* __restrict__ sigma,
                  const float* __restrict__ epsilon,
                  float* __restrict__ rec,     // 8 floats per atom
                  int n_atoms)
{
    const int tid    = blockIdx.x * blockDim.x + threadIdx.x;
    const int stride = gridDim.x * blockDim.x;
    for (int a = tid; a < n_atoms; a += stride) {
        float4 r0;
        r0.x = coords[3*a + 0];
        r0.y = coords[3*a + 1];
        r0.z = coords[3*a + 2];
        r0.w = 0.5f * sigma[a];
        ((float4*)rec)[2*a] = r0;
        rec[8*a + 4] = sqrtf(epsilon[a]);
    }
}

// ---------------------------------------------------------------------------
// Main pair kernel: NT-hinted pair stream, 2-way unrolled grid-stride for MLP.
// ---------------------------------------------------------------------------
__global__ void __launch_bounds__(256)
lj_pair_energy_packed(const float* __restrict__ rec,      // 8 floats/atom
                      const long long* __restrict__ pairs,
                      const float* __restrict__ box,
                      const int* __restrict__ cutoff_p,
                      float* __restrict__ out,
                      int n_pairs)
{
    const BoxMats m = load_box(box);
    const float cutoff = (float)cutoff_p[0];

    const v2ll* __restrict__ pp = (const v2ll*)pairs;

    float acc0 = 0.0f, acc1 = 0.0f;
    const int tid    = blockIdx.x * blockDim.x + threadIdx.x;
    const int stride = gridDim.x * blockDim.x;
    const int stride2 = stride * 2;

    int p = tid;
    // 2 pairs in flight per wave per iteration: 2 NT pair loads + 8 gathers
    for (; p + stride < n_pairs; p += stride2) {
        __builtin_prefetch(&pp[(size_t)p + (size_t)stride2], 0, 0);
        __builtin_prefetch(&pp[(size_t)p + (size_t)stride2 + (size_t)stride], 0, 0);

        const v2ll pr0 = __builtin_nontemporal_load(&pp[(size_t)p]);           // b128 NT
        const v2ll pr1 = __builtin_nontemporal_load(&pp[(size_t)p + (size_t)stride]);

        acc0 += pair_energy_packed(rec, pr0.x, pr0.y, m, cutoff);
        acc1 += pair_energy_packed(rec, pr1.x, pr1.y, m, cutoff);
    }
    if (p < n_pairs) {
        const v2ll pr0 = __builtin_nontemporal_load(&pp[(size_t)p]);
        acc0 += pair_energy_packed(rec, pr0.x, pr0.y, m, cutoff);
    }

    block_reduce_add(acc0 + acc1, out);
}

// ---------------------------------------------------------------------------
// Fallback: direct gathers (used only if ws_size is too small to pack).
// ---------------------------------------------------------------------------
__global__ void __launch_bounds__(256)
lj_pair_energy_direct(const float* __restrict__ coords,
                      const long long* __restrict__ pairs,
                      const float* __restrict__ box,
                      const float* __restrict__ sigma,
                      const float* __restrict__ epsilon,
                      const int* __restrict__ cutoff_p,
                      float* __restrict__ out,
                      int n_pairs)
{
    const BoxMats m = load_box(box);
    const float cutoff = (float)cutoff_p[0];

    const LLPair* __restrict__ pp = (const LLPair*)pairs;

    float acc = 0.0f;
    const int tid    = blockIdx.x * blockDim.x + threadIdx.x;
    const int stride = gridDim.x * blockDim.x;

    for (int p = tid; p < n_pairs; p += stride) {
        __builtin_prefetch(&pp[(size_t)p + (size_t)stride], 0, 0);

        const LLPair pr = pp[p];
        const size_t ia = (size_t)pr.i, ja = (size_t)pr.j;

        const float xi = coords[3*ia + 0], yi = coords[3*ia + 1], zi = coords[3*ia + 2];
        const float xj = coords[3*ja + 0], yj = coords[3*ja + 1], zj = coords[3*ja + 2];

        const float dx = xi - xj, dy = yi - yj, dz = zi - zj;

        float sx = dx*m.i00 + dy*m.i10 + dz*m.i20;
        float sy = dx*m.i01 + dy*m.i11 + dz*m.i21;
        float sz = dx*m.i02 + dy*m.i12 + dz*m.i22;
        sx -= floorf(sx + 0.5f);
        sy -= floorf(sy + 0.5f);
        sz -= floorf(sz + 0.5f);

        const float px = sx*m.b00 + sy*m.b10 + sz*m.b20;
        const float py = sx*m.b01 + sy*m.b11 + sz*m.b21;
        const float pz = sx*m.b02 + sy*m.b12 + sz*m.b22;

        const float d2 = px*px + py*py + pz*pz;
        const float d  = sqrtf(d2);
        const float dsafe = (d > 0.0f) ? d : 1.0f;

        const float sg = 0.5f * (sigma[ia] + sigma[ja]);
        const float ep = sqrtf(epsilon[ia] * epsilon[ja]);

        const float s  = sg / dsafe;
        const float s2 = s * s;
        const float t  = s2 * s2 * s2;
        const float e  = 4.0f * ep * t * (t - 1.0f);

        acc += (d <= cutoff) ? e : 0.0f;
    }

    block_reduce_add(acc, out);
}

extern "C" void kernel_launch(void* const* d_in, const int* in_sizes, int n_in,
                              void* d_out, int out_size, void* d_ws, size_t ws_size,
                              hipStream_t stream) {
    const float*     coords  = (const float*)d_in[0];
    const long long* pairs   = (const long long*)d_in[1];
    const float*     box     = (const float*)d_in[2];
    const float*     sigma   = (const float*)d_in[3];
    const float*     epsilon = (const float*)d_in[4];
    const int*       cutoff  = (const int*)d_in[5];

    const int n_atoms = in_sizes[0] / 3;
    const int n_pairs = in_sizes[1] / 2;

    hipMemsetAsync(d_out, 0, sizeof(float) * (out_size > 0 ? out_size : 1), stream);

    const int threads = 256;           // 8 wave32 waves per block
    const int blocks  = 2048;          // grid-stride: ~6 double-iterations/thread

    const size_t ws_needed = (size_t)n_atoms * 8 * sizeof(float);
    if (ws_size >= ws_needed) {
        float* rec = (float*)d_ws;
        const int pblocks = (n_atoms + threads - 1) / threads;
        pack_atoms_kernel<<<pblocks, threads, 0, stream>>>(
            coords, sigma, epsilon, rec, n_atoms);
        lj_pair_energy_packed<<<blocks, threads, 0, stream>>>(
            rec, pairs, box, cutoff, (float*)d_out, n_pairs);
    } else {
        lj_pair_energy_direct<<<blocks, threads, 0, stream>>>(
            coords, pairs, box, sigma, epsilon, cutoff, (float*)d_out, n_pairs);
    }
}